// MipNeRFRenderer_47536698032154
// MI455X (gfx1250) — compile-verified
//
#include <hip/hip_runtime.h>

// ---------------------------------------------------------------------------
// MipNeRF hierarchical sampling + volumetric rendering, MI455X (gfx1250).
//
// One wave32 per ray, 8 rays per 256-thread block.
//   phase 0a: block-wide TDM tensor_load_to_lds stages the 8 contiguous
//             t_vals rows (2 KB) in one DMA descriptor (TENSORcnt path),
//             issued by wave 0, published via one s_barrier.
//   phase 0b: per-wave global_load_async_to_lds_b32 stages the weights row
//             (ASYNCcnt path), overlapped with u loads + global_prefetch of
//             the density/color rows.
//   phase 1:  wave-scan (shfl_up) builds normalized 63-entry CDF in LDS
//   phase 2:  4 binary searches / lane -> 128 importance samples (registers)
//   phase 3:  register bitonic sort of 256 (192 real + 64 +inf pad),
//             8 elems/lane, cross-lane stages via shfl_xor
//   phase 4:  dists from shuffles of sorted regs, exp/alpha, segmented
//             shfl_up product scan for exclusive transmittance, coalesced w
//             store, shfl_xor reductions for rgb/depth/acc.
//
// Traffic ~0.3 GB -> ~13 us at 23.3 TB/s; bitonic keeps sort VALU ~1.3K
// wave-instr/ray so we stay near the bandwidth bound. No matmul structure
// exists in this workload, so no WMMA by design.
// ---------------------------------------------------------------------------

namespace {
constexpr int SC   = 64;    // coarse samples per ray
constexpr int NI   = 128;   // importance samples per ray
constexpr int ST   = 192;   // total samples per ray
constexpr int NB   = 62;    // interior coarse weights (drop first/last)
constexpr int NCDF = 63;    // cdf entries = NB + 1
constexpr int WPB  = 8;     // waves (rays) per block
}

typedef unsigned int v4u __attribute__((ext_vector_type(4)));
typedef unsigned int v8u __attribute__((ext_vector_type(8)));

__device__ __forceinline__ unsigned lds_lo32(const void* p) {
  // flat LDS pointer: low 32 bits are the workgroup-relative LDS byte offset
  return (unsigned)(unsigned long long)p;
}

__device__ __forceinline__ unsigned long long wave_uniform_ptr(const void* p) {
  unsigned long long x = (unsigned long long)p;
  unsigned lo = (unsigned)__builtin_amdgcn_readfirstlane((int)(unsigned)x);
  unsigned hi = (unsigned)__builtin_amdgcn_readfirstlane((int)(unsigned)(x >> 32));
  return ((unsigned long long)hi << 32) | lo;
}

__device__ __forceinline__ void async_g2l_b32(unsigned lds, unsigned voff,
                                              unsigned long long sbase) {
  // GVS mode: per-lane LDS dest VGPR, per-lane 32-bit offset VGPR, SGPR base
  asm volatile("global_load_async_to_lds_b32 %0, %1, %2"
               :
               : "v"(lds), "v"(voff), "s"(sbase)
               : "memory");
}

__device__ __forceinline__ void wait_asynccnt0() {
  asm volatile("s_wait_asynccnt 0x0" ::: "memory");
}

// One-descriptor 1-D TDM load: nelem f32 from gaddr -> LDS ldsoff.
// D# layout per CDNA5 ISA (groups 2/3 disabled).
__device__ __forceinline__ void tdm_load_1d_f32(unsigned ldsoff,
                                                unsigned long long gaddr,
                                                unsigned nelem) {
  v4u g0;
  g0.x = 1u;                                  // count=1, user mode
  g0.y = ldsoff;                              // lds_addr (bytes)
  g0.z = (unsigned)gaddr;                     // global_addr[31:0]
  g0.w = ((unsigned)(gaddr >> 32) & 0x01FFFFFFu) | 0x80000000u; // addr[56:32] | type=2

  v8u g1;
  g1.s0 = 0x00020000u;                        // wg_mask=0, data_size=2 (4B)
  g1.s1 = (nelem & 0xFFFFu) << 16;            // tensor_dim0[15:0]
  g1.s2 = (nelem >> 16) | (1u << 16);         // tensor_dim0[31:16] | tensor_dim1=1
  g1.s3 = (nelem & 0xFFFFu) << 16;            // tile_dim0
  g1.s4 = 1u;                                 // tile_dim1=1, tile_dim2=0
  g1.s5 = nelem;                              // tensor_dim0_stride[31:0]
  g1.s6 = 0u;                                 // stride hi | tensor_dim1_stride lo
  g1.s7 = 0u;

  asm volatile("tensor_load_to_lds %0, %1"
               :
               : "s"(g0), "s"(g1)
               : "memory");
}

__global__ __launch_bounds__(256) void
mipnerf_render_gfx1250(const float* __restrict__ t_vals,
                       const float* __restrict__ weights,
                       const float* __restrict__ u,
                       const float* __restrict__ dens,
                       const float* __restrict__ cols,
                       float* __restrict__ out, int N) {
  __shared__ float sh_tv [WPB][SC];           // filled by one block-wide TDM op
  __shared__ float sh_w  [WPB][SC];           // filled by per-wave async loads
  __shared__ float sh_cdf[WPB][NCDF + 1];

  const int lane = (int)(threadIdx.x & 31u);
  const int wv   = (int)(threadIdx.x >> 5);
  const int ray0 = (int)blockIdx.x * WPB;
  const int ray  = ray0 + wv;
  const bool active = (ray < N);              // uniform per wave

  float uu[4];
  const float* gd = dens + (size_t)ray * ST;
  const float* gc = cols + (size_t)ray * (size_t)ST * 3;

  if (active) {
    // ---- phase 0b: per-wave async stage of the weights row (ASYNCcnt) ----
    const unsigned long long gw = wave_uniform_ptr(weights + (size_t)ray * SC);
    const unsigned lw = lds_lo32(&sh_w[wv][lane]);
    const unsigned vo = (unsigned)lane * 4u;
    async_g2l_b32(lw,        vo,        gw);
    async_g2l_b32(lw + 128u, vo + 128u, gw);

    // prefetch the fine-pass payload rows (consumed after sampling+sort)
    __builtin_prefetch(gd + lane * 6, 0, 0);   // 6 x f32 per lane = full row
    __builtin_prefetch(gc + lane * 18, 0, 0);  // 18 x f32 per lane = full row

    // load u while the async copy / TDM are in flight
    const float* gu = u + (size_t)ray * NI;
#pragma unroll
    for (int q = 0; q < 4; ++q) uu[q] = gu[lane + 32 * q];
  }

  // ---- phase 0a: one TDM descriptor stages all 8 t_vals rows (TENSORcnt) --
  if (wv == 0) {
    unsigned rows = (unsigned)((N - ray0) < WPB ? (N - ray0) : WPB);
    tdm_load_1d_f32(lds_lo32(&sh_tv[0][0]),
                    (unsigned long long)(t_vals + (size_t)ray0 * SC),
                    rows * (unsigned)SC);
    __builtin_amdgcn_s_wait_tensorcnt(0);
  }
  __syncthreads();                             // publish sh_tv to all waves
  if (!active) return;

  wait_asynccnt0();                            // own wave's weights row ready

  // ---- phase 1: normalized CDF over interior weights ----------------------
  // lane i owns interior indices k0=2i, k1=2i+1 (k < 62)
  const int k0 = 2 * lane, k1 = 2 * lane + 1;
  float a = (k0 < NB) ? (sh_w[wv][1 + k0] + 1e-5f) : 0.0f;
  float b = (k1 < NB) ? (sh_w[wv][1 + k1] + 1e-5f) : 0.0f;
  float pair = a + b;
  float s = pair;                       // inclusive scan of pair sums
#pragma unroll
  for (int d = 1; d < 32; d <<= 1) {
    float t = __shfl_up(s, d, 32);
    if (lane >= d) s += t;
  }
  float total = __shfl(s, 31, 32);
  float excl  = __shfl_up(s, 1, 32);
  if (lane == 0) excl = 0.0f;
  const float inv = 1.0f / total;
  if (lane == 0) sh_cdf[wv][0] = 0.0f;
  if (k0 < NB) sh_cdf[wv][1 + k0] = (excl + a) * inv;
  if (k1 < NB) sh_cdf[wv][1 + k1] = (excl + pair) * inv;

  // ---- phase 2: inverse-CDF sampling, 4 samples per lane ------------------
  // sort registers: element e = (r<<5)|lane; v[0..1]=t_vals, v[2..5]=samples
  const float* cdfp = &sh_cdf[wv][0];
  const float* tvp  = &sh_tv[wv][0];
  float v[8];
  v[0] = tvp[lane];
  v[1] = tvp[lane + 32];
  v[6] = __builtin_inff();
  v[7] = __builtin_inff();

#pragma unroll
  for (int q = 0; q < 4; ++q) {
    const float uq = uu[q];
    int lo = 0, hi = NCDF;              // upper_bound over cdf[0..62]
#pragma unroll
    for (int it = 0; it < 6; ++it) {
      if (lo < hi) {
        int m = (lo + hi) >> 1;         // m <= 62 always
        if (cdfp[m] <= uq) lo = m + 1; else hi = m;
      }
    }
    int inds  = lo;                                   // in [0, 63]
    int below = inds - 1; below = below < 0 ? 0 : (below > 62 ? 62 : below);
    int above = inds;     above = above > 62 ? 62 : above;
    float c0 = cdfp[below], c1 = cdfp[above];
    float b0 = tvp[below],  b1 = tvp[above];
    float den = c1 - c0;
    if (den < 1e-5f) den = 1.0f;
    float tt = (uq - c0) / den;
    v[2 + q] = b0 + tt * (b1 - b0);
  }

  // ---- phase 3: bitonic sort of 256 (ascending), 8 elems/lane -------------
#pragma unroll
  for (unsigned k = 2; k <= 256; k <<= 1) {
#pragma unroll
    for (unsigned j = k >> 1; j > 0; j >>= 1) {
      if (j >= 32u) {
        const unsigned jr = j >> 5;                   // intra-lane, r-bit stride
#pragma unroll
        for (unsigned r = 0; r < 8; ++r) {
          if ((r & jr) == 0) {
            const unsigned r2 = r | jr;
            const bool asc = (((r << 5) & k) == 0);   // k >= 64 here
            float mn = fminf(v[r], v[r2]);
            float mx = fmaxf(v[r], v[r2]);
            v[r]  = asc ? mn : mx;
            v[r2] = asc ? mx : mn;
          }
        }
      } else {                                        // cross-lane stride
#pragma unroll
        for (unsigned r = 0; r < 8; ++r) {
          float x = __shfl_xor(v[r], (int)j, 32);
          const unsigned i = (r << 5) | (unsigned)lane;
          const bool asc   = ((i & k) == 0);
          const bool lower = (((unsigned)lane & j) == 0);
          v[r] = (lower == asc) ? fminf(v[r], x) : fmaxf(v[r], x);
        }
      }
    }
  }
  // sorted t: position p = r*32 + lane holds v[r]; p in [0,191] real

  // ---- phase 4: volumetric rendering --------------------------------------
  float* orgb = out;
  float* odep = out + 3 * (size_t)N;
  float* oacc = out + 4 * (size_t)N;
  float* ow   = out + 5 * (size_t)N + (size_t)ray * ST;

  float rgb0 = 0.f, rgb1 = 0.f, rgb2 = 0.f, dep = 0.f, acc = 0.f;
  float carry = 1.0f;                                 // running transmittance

#pragma unroll
  for (int r = 0; r < 6; ++r) {
    const float t  = v[r];
    float tn  = __shfl_down(v[r], 1, 32);             // next within chunk
    float tfn = __shfl(v[(r + 1) & 7], 0, 32);        // first of next chunk
    if (lane == 31) tn = tfn;
    float dist = (r == 5 && lane == 31) ? 1e10f : (tn - t);

    const int p = r * 32 + lane;
    float sg = gd[p];
    sg = sg > 0.0f ? sg : 0.0f;                       // relu
    float alpha = 1.0f - expf(-sg * dist);
    float f = 1.0f - alpha + 1e-10f;

    // inclusive product scan of f across the wave
    float ss = f;
#pragma unroll
    for (int d = 1; d < 32; d <<= 1) {
      float tt = __shfl_up(ss, d, 32);
      if (lane >= d) ss *= tt;
    }
    float ex = __shfl_up(ss, 1, 32);
    if (lane == 0) ex = 1.0f;
    float trans = carry * ex;                         // exclusive transmittance
    carry *= __shfl(ss, 31, 32);

    float w = alpha * trans;
    ow[p] = w;                                        // coalesced b32 store

    float c0 = gc[p * 3 + 0], c1 = gc[p * 3 + 1], c2 = gc[p * 3 + 2];
    rgb0 += w * c0;
    rgb1 += w * c1;
    rgb2 += w * c2;
    dep  += w * t;
    acc  += w;
  }

  // wave reduction of the 5 accumulators
#pragma unroll
  for (int m = 16; m >= 1; m >>= 1) {
    rgb0 += __shfl_xor(rgb0, m, 32);
    rgb1 += __shfl_xor(rgb1, m, 32);
    rgb2 += __shfl_xor(rgb2, m, 32);
    dep  += __shfl_xor(dep,  m, 32);
    acc  += __shfl_xor(acc,  m, 32);
  }
  if (lane == 0) {
    orgb[(size_t)ray * 3 + 0] = rgb0;
    orgb[(size_t)ray * 3 + 1] = rgb1;
    orgb[(size_t)ray * 3 + 2] = rgb2;
    odep[ray] = dep;
    oacc[ray] = acc;
  }
}

extern "C" void kernel_launch(void* const* d_in, const int* in_sizes, int n_in,
                              void* d_out, int out_size, void* d_ws, size_t ws_size,
                              hipStream_t stream) {
  (void)n_in; (void)out_size; (void)d_ws; (void)ws_size;
  const float* t_vals  = (const float*)d_in[0];
  const float* weights = (const float*)d_in[1];
  const float* u       = (const float*)d_in[2];
  const float* dens    = (const float*)d_in[3];
  const float* cols    = (const float*)d_in[4];
  float* out = (float*)d_out;

  const int N = in_sizes[0] / SC;                     // 65536
  const int blocks = (N + WPB - 1) / WPB;             // 8192
  mipnerf_render_gfx1250<<<dim3(blocks), dim3(256), 0, stream>>>(
      t_vals, weights, u, dens, cols, out, N);
}